// Complex_LayerNorm_14645838480097
// MI455X (gfx1250) — compile-verified
//
#include <hip/hip_runtime.h>
#include <hip/hip_bf16.h>

#define CLN_D   1024      // embed dim (fixed by reference)
#define CLN_NT  256       // threads per block (8 wave32s)
#define CLN_NW  (CLN_NT / 32)

typedef float v4f __attribute__((ext_vector_type(4)));

// ---------------------------------------------------------------------------
// Kernel 1: per-feature recoloring matrix, computed once per launch.
//   wr = w0^2 ; wi = w1^2 ; wc = (sigmoid(w2)-0.5)*2*sqrt(wr*wi)
//   W  = [[wr,wc],[wc,wi]] ; Wsqrt = [[wr+s, wc],[wc, wi+s]]/t
//   with s = sqrt(wr*wi - wc^2), t = sqrt(wr+wi+2s)
// Stores g00,g01,g11 (symmetric) into workspace (L2-resident table).
// ---------------------------------------------------------------------------
__global__ __launch_bounds__(CLN_NT) void cln_prep(const float* __restrict__ w,
                                                   float* __restrict__ g00,
                                                   float* __restrict__ g01,
                                                   float* __restrict__ g11,
                                                   int Dn) {
    int d = blockIdx.x * blockDim.x + threadIdx.x;
    if (d >= Dn) return;
    float w0 = w[d * 3 + 0];
    float w1 = w[d * 3 + 1];
    float w2 = w[d * 3 + 2];
    float wr = w0 * w0;
    float wi = w1 * w1;
    float sig = 1.0f / (1.0f + __expf(-w2));
    float wc = (sig - 0.5f) * 2.0f * sqrtf(wr * wi);
    float s  = sqrtf(wr * wi - wc * wc);
    float t  = sqrtf(wr + wi + 2.0f * s);
    float rt = 1.0f / t;
    g00[d] = (wr + s) * rt;
    g01[d] = wc * rt;
    g11[d] = (wi + s) * rt;
}

// ---------------------------------------------------------------------------
// Elementwise apply: whiten (2x2) then recolor (2x2) + bias -> complex pair.
// ---------------------------------------------------------------------------
__device__ __forceinline__ float2 cln_apply(float xr, float xi,
                                            float mur, float mui,
                                            float c00, float c01, float c11,
                                            float g00, float g01, float g11,
                                            float br, float bi) {
    float xrc = xr - mur;
    float xic = xi - mui;
    float yr = fmaf(c00, xrc, c01 * xic);
    float yi = fmaf(c01, xrc, c11 * xic);
    float zr = fmaf(g00, yr, fmaf(g01, yi, br));
    float zi = fmaf(g01, yr, fmaf(g11, yi, bi));
    return make_float2(zr, zi);
}

// ---------------------------------------------------------------------------
// Kernel 2: one block per row. Async-copy the row (xr 4KB + xi 4KB) into LDS
// via the CDNA5 async global->LDS path with non-temporal policy (stream-once
// data must not churn the 192MB L2), reduce 5 moments with wave32 shuffles,
// compute the whitening matrix inv(sqrt2x2(C)) in closed form, broadcast,
// apply, and store interleaved complex64 with non-temporal 128-bit stores.
// ---------------------------------------------------------------------------
__global__ __launch_bounds__(CLN_NT) void cln_main(const float* __restrict__ xr,
                                                   const float* __restrict__ xi,
                                                   const float* __restrict__ g00,
                                                   const float* __restrict__ g01,
                                                   const float* __restrict__ g11,
                                                   const float* __restrict__ br,
                                                   const float* __restrict__ bi,
                                                   float* __restrict__ out) {
    __shared__ __align__(16) float s_xr[CLN_D];
    __shared__ __align__(16) float s_xi[CLN_D];
    __shared__ float s_red[CLN_NW * 5];
    __shared__ float s_bc[5];

    const int tid = threadIdx.x;
    const int d0  = tid * 4;                       // 4 features per thread
    const long long base = (long long)blockIdx.x * CLN_D;

    // ---- async global -> LDS copy of this row (128b per lane, NT policy) ----
    const float* gr = xr + base + d0;
    const float* gi = xi + base + d0;
    unsigned lr = (unsigned)(unsigned long long)(&s_xr[d0]);
    unsigned li = (unsigned)(unsigned long long)(&s_xi[d0]);
    asm volatile("global_load_async_to_lds_b128 %0, %1, off th:TH_LOAD_NT"
                 :: "v"(lr), "v"(gr) : "memory");
    asm volatile("global_load_async_to_lds_b128 %0, %1, off th:TH_LOAD_NT"
                 :: "v"(li), "v"(gi) : "memory");
    asm volatile("s_wait_asynccnt 0" ::: "memory");
    __syncthreads();

    const float4 vr = *reinterpret_cast<const float4*>(&s_xr[d0]);
    const float4 vi = *reinterpret_cast<const float4*>(&s_xi[d0]);

    // ---- per-thread partial moments ----
    float sr  = (vr.x + vr.y) + (vr.z + vr.w);
    float si  = (vi.x + vi.y) + (vi.z + vi.w);
    float srr = fmaf(vr.x, vr.x, fmaf(vr.y, vr.y, fmaf(vr.z, vr.z, vr.w * vr.w)));
    float sii = fmaf(vi.x, vi.x, fmaf(vi.y, vi.y, fmaf(vi.z, vi.z, vi.w * vi.w)));
    float sri = fmaf(vr.x, vi.x, fmaf(vr.y, vi.y, fmaf(vr.z, vi.z, vr.w * vi.w)));

    // ---- wave32 shuffle reduction ----
#pragma unroll
    for (int off = 16; off > 0; off >>= 1) {
        sr  += __shfl_xor(sr,  off, 32);
        si  += __shfl_xor(si,  off, 32);
        srr += __shfl_xor(srr, off, 32);
        sii += __shfl_xor(sii, off, 32);
        sri += __shfl_xor(sri, off, 32);
    }
    if ((tid & 31) == 0) {
        int w = tid >> 5;
        s_red[0 * CLN_NW + w] = sr;
        s_red[1 * CLN_NW + w] = si;
        s_red[2 * CLN_NW + w] = srr;
        s_red[3 * CLN_NW + w] = sii;
        s_red[4 * CLN_NW + w] = sri;
    }
    __syncthreads();

    // ---- final combine + closed-form inv(sqrt2x2(cov)) in one lane ----
    if (tid == 0) {
        float Sr = 0.f, Si = 0.f, Srr = 0.f, Sii = 0.f, Sri = 0.f;
#pragma unroll
        for (int w = 0; w < CLN_NW; ++w) {
            Sr  += s_red[0 * CLN_NW + w];
            Si  += s_red[1 * CLN_NW + w];
            Srr += s_red[2 * CLN_NW + w];
            Sii += s_red[3 * CLN_NW + w];
            Sri += s_red[4 * CLN_NW + w];
        }
        const float n    = (float)CLN_D;
        const float rn   = 1.0f / n;
        const float rnm1 = 1.0f / (n - 1.0f);
        const float mur = Sr * rn;
        const float mui = Si * rn;
        const float a   = (Srr - n * mur * mur) * rnm1;   // var_r (unbiased)
        const float dv  = (Sii - n * mui * mui) * rnm1;   // var_i (unbiased)
        const float b   = Sri * rn - mur * mui;           // biased cross-cov
        // sqrt2x2 of [[a,b],[b,dv]] then inv2x2, folded:
        const float s   = sqrtf(a * dv - b * b);
        const float t   = sqrtf(a + dv + 2.0f * s);
        const float m00 = a + s;
        const float m11 = dv + s;
        const float k   = t / (m00 * m11 - b * b);
        s_bc[0] =  m11 * k;   // inv[0,0]
        s_bc[1] = -b   * k;   // inv[0,1] == inv[1,0]
        s_bc[2] =  m00 * k;   // inv[1,1]
        s_bc[3] = mur;
        s_bc[4] = mui;
    }
    __syncthreads();

    const float c00 = s_bc[0], c01 = s_bc[1], c11 = s_bc[2];
    const float mur = s_bc[3], mui = s_bc[4];

    // per-feature recolor matrix + bias: default (RT) policy -> stays in L2,
    // reused by all 32768 row-blocks.
    const float4 G00 = *reinterpret_cast<const float4*>(g00 + d0);
    const float4 G01 = *reinterpret_cast<const float4*>(g01 + d0);
    const float4 G11 = *reinterpret_cast<const float4*>(g11 + d0);
    const float4 BR  = *reinterpret_cast<const float4*>(br  + d0);
    const float4 BI  = *reinterpret_cast<const float4*>(bi  + d0);

    const float2 z0 = cln_apply(vr.x, vi.x, mur, mui, c00, c01, c11, G00.x, G01.x, G11.x, BR.x, BI.x);
    const float2 z1 = cln_apply(vr.y, vi.y, mur, mui, c00, c01, c11, G00.y, G01.y, G11.y, BR.y, BI.y);
    const float2 z2 = cln_apply(vr.z, vi.z, mur, mui, c00, c01, c11, G00.z, G01.z, G11.z, BR.z, BI.z);
    const float2 z3 = cln_apply(vr.w, vi.w, mur, mui, c00, c01, c11, G00.w, G01.w, G11.w, BR.w, BI.w);

    // interleaved complex64 output: write-once data -> non-temporal stores
    float* op = out + 2 * (base + d0);
    v4f o01 = { z0.x, z0.y, z1.x, z1.y };
    v4f o23 = { z2.x, z2.y, z3.x, z3.y };
    __builtin_nontemporal_store(o01, reinterpret_cast<v4f*>(op));
    __builtin_nontemporal_store(o23, reinterpret_cast<v4f*>(op + 4));
}

// ---------------------------------------------------------------------------
// Launch wrapper
// inputs: 0=x_real [B,S,D] f32, 1=x_imag [B,S,D] f32, 2=weights [D,3,1] f32,
//         3=bias_real [D] f32, 4=bias_imag [D] f32
// output: complex64 [B,S,D] -> float* interleaved (re,im)
// ---------------------------------------------------------------------------
extern "C" void kernel_launch(void* const* d_in, const int* in_sizes, int n_in,
                              void* d_out, int out_size, void* d_ws, size_t ws_size,
                              hipStream_t stream) {
    const float* xr = (const float*)d_in[0];
    const float* xi = (const float*)d_in[1];
    const float* w  = (const float*)d_in[2];
    const float* br = (const float*)d_in[3];
    const float* bi = (const float*)d_in[4];
    float* out = (float*)d_out;
    float* ws  = (float*)d_ws;

    const int Dn = in_sizes[3];                          // = 1024
    const long long rows = (long long)in_sizes[0] / Dn;  // B*S = 32768

    float* g00 = ws;
    float* g01 = ws + Dn;
    float* g11 = ws + 2 * Dn;

    cln_prep<<<(Dn + CLN_NT - 1) / CLN_NT, CLN_NT, 0, stream>>>(w, g00, g01, g11, Dn);
    cln_main<<<(unsigned)rows, CLN_NT, 0, stream>>>(xr, xi, g00, g01, g11, br, bi, out);
}